// ModifiedLogicSegLoss_17265768530476
// MI455X (gfx1250) — compile-verified
//
#include <hip/hip_runtime.h>
#include <hip/hip_bf16.h>

typedef __attribute__((ext_vector_type(2))) float v2f;
typedef __attribute__((ext_vector_type(8))) float v8f;

#define H_LEVELS 4
#define NCLS 128
#define LOG_CLAMP_MAG 100.0f

// bce(x, t) with t in {0,1}:
//   t=1 -> min(-log(sigmoid(x)), 100) = min(softplus(-x), 100)
//   t=0 -> min(-log1p(-sigmoid(x)), 100) = min(softplus(x), 100)
__device__ __forceinline__ float bce_elem(float x, float t) {
    float z  = (t > 0.5f) ? -x : x;          // v_cndmask
    float e  = __expf(z);                    // v_exp_f32 (scaled)
    float sp = __logf(1.0f + e);             // softplus; inf -> clamped below
    return fminf(sp, LOG_CLAMP_MAG);
}

// Full 32-lane sum via one V_WMMA_F32_16X16X4_F32.
// A (16x4): lanes 0-15 supply K=0,1 from vgpr0/1; lanes 16-31 supply K=2,3.
// With a0=s, a1=0 and B=ones: D[m,n] = s_m + s_{m+16} (independent of n).
// D vgpr j: lanes 0-15 hold M=j, lanes 16-31 hold M=j+8, so a local sum of the
// 8 D vgprs gives sum(r_0..r_7) in lanes<16 and sum(r_8..r_15) in lanes>=16;
// one shfl_xor(16) completes the reduction.
__device__ __forceinline__ float wave_sum_wmma(float s) {
    v2f a; a.x = s;    a.y = 0.0f;
    v2f b; b.x = 1.0f; b.y = 1.0f;
    v8f c = {0.0f, 0.0f, 0.0f, 0.0f, 0.0f, 0.0f, 0.0f, 0.0f};
    c = __builtin_amdgcn_wmma_f32_16x16x4_f32(
            /*neg_a=*/false, a, /*neg_b=*/false, b,
            /*c_mod=*/(short)0, c, /*reuse_a=*/false, /*reuse_b=*/false);
    float r = ((c[0] + c[1]) + (c[2] + c[3])) + ((c[4] + c[5]) + (c[6] + c[7]));
    r += __shfl_xor(r, 16, 32);
    return r;
}

__global__ void __launch_bounds__(256)
lseg_bce_partial(const float* __restrict__ y_pred,
                 const float* __restrict__ y_true,
                 const float* __restrict__ La,
                 const float* __restrict__ lam,
                 float* __restrict__ partials,
                 int n_elems) {
    __shared__ float wcls[NCLS];   // per-class weight = lam[level(c)] / 128
    __shared__ float wsum[8];
    const int tid = threadIdx.x;

    if (tid < NCLS) {
        float w = 0.0f;
#pragma unroll
        for (int l = 0; l < H_LEVELS; ++l)
            w += La[l * NCLS + tid] * lam[l];
        wcls[tid] = w * (1.0f / (float)NCLS);
    }
    __syncthreads();

    const int     n4      = n_elems >> 2;                 // float4 count
    const int     gtid    = blockIdx.x * 256 + tid;
    const int     gstride = gridDim.x * 256;
    const float4* xp      = (const float4*)y_pred;
    const float4* tp      = (const float4*)y_true;

    float acc = 0.0f;
    for (int i = gtid; i < n4; i += gstride) {
        float4 x = xp[i];                                 // global_load_b128
        float4 t = tp[i];                                 // global_load_b128
        // classes of this float4: ((i*4) mod 128) .. +3  ==  ((i & 31) << 2)
        const float4 w = *(const float4*)&wcls[(i & 31) << 2];  // ds_load_b128
        acc += w.x * bce_elem(x.x, t.x);
        acc += w.y * bce_elem(x.y, t.y);
        acc += w.z * bce_elem(x.z, t.z);
        acc += w.w * bce_elem(x.w, t.w);
    }

    const float wsr = wave_sum_wmma(acc);                 // uniform control flow, EXEC all-1s
    if ((tid & 31) == 0) wsum[tid >> 5] = wsr;
    __syncthreads();
    if (tid == 0) {
        float r = 0.0f;
#pragma unroll
        for (int w = 0; w < 8; ++w) r += wsum[w];
        partials[blockIdx.x] = r;                         // deterministic, no float atomics
    }
}

__global__ void __launch_bounds__(256)
lseg_bce_final(const float* __restrict__ partials,
               float* __restrict__ out,
               int n) {
    __shared__ float wsum[8];
    const int tid = threadIdx.x;
    float acc = 0.0f;
    for (int i = tid; i < n; i += 256)                    // fixed ascending order
        acc += partials[i];
    const float wsr = wave_sum_wmma(acc);
    if ((tid & 31) == 0) wsum[tid >> 5] = wsr;
    __syncthreads();
    if (tid == 0) {
        float r = 0.0f;
#pragma unroll
        for (int w = 0; w < 8; ++w) r += wsum[w];
        out[0] = r;                                       // ALPHA_BCE == 1.0
    }
}

extern "C" void kernel_launch(void* const* d_in, const int* in_sizes, int n_in,
                              void* d_out, int out_size, void* d_ws, size_t ws_size,
                              hipStream_t stream) {
    const float* y_pred = (const float*)d_in[0];   // [B, C] f32
    const float* y_true = (const float*)d_in[1];   // [B, C] f32 (0/1)
    const float* La     = (const float*)d_in[2];   // [4, 128] f32
    const float* lam    = (const float*)d_in[3];   // [4] f32
    float*       parts  = (float*)d_ws;            // 4096 floats = 16 KB scratch

    const int n  = in_sizes[0];                    // B*C = 33,554,432
    const int NB = 4096;                           // 1M threads -> exactly 8 float4 iters each

    lseg_bce_partial<<<NB, 256, 0, stream>>>(y_pred, y_true, La, lam, parts, n);
    lseg_bce_final<<<1, 256, 0, stream>>>(parts, (float*)d_out, NB);
}